// GraphAttentionNetwork_66898410602986
// MI455X (gfx1250) — compile-verified
//
#include <hip/hip_runtime.h>
#include <math.h>

typedef __attribute__((ext_vector_type(2))) float v2f;
typedef __attribute__((ext_vector_type(8))) float v8f;

#define BATCH 8
#define GAT_ALPHA 0.2f

// ---------------------------------------------------------------------------
// Kernel 1: L[b,n,d] = sum_f x[b,f,n]*W[d,f]   (+ b_lin folded into L)
//           R[b,n,d] = sum_f x[b,f,n]*W[d,256+f]
// One wave per 16(n) x 64(d) strip: 4 independent WMMA accumulator chains
// sharing one A-fragment (4x A reuse, overlapped WMMA pipelines).
// ---------------------------------------------------------------------------
__global__ __launch_bounds__(32) void gat_proj(const float* __restrict__ x,
                                               const float* __restrict__ W,
                                               const float* __restrict__ b_lin,
                                               float* __restrict__ Lbuf,
                                               float* __restrict__ Rbuf) {
  const int b     = blockIdx.z;
  const int ntile = blockIdx.y;        // 0..15 : 16-row tile over n
  const int strip = blockIdx.x;        // 0..7  : <4 -> L, else R; 64 d-cols each
  const int lane  = threadIdx.x;
  const int m     = lane & 15;         // M index / B-N index within tile
  const int kh    = lane >> 4;         // K-half select for A/B fragments

  const bool isR  = strip >= 4;
  const int d0    = (strip & 3) * 64;
  const int n0    = ntile * 16;
  const int wf0   = isR ? 256 : 0;     // W column offset for reduction axis

  const float* xb = x + (size_t)b * 65536;

  v8f c0 = {}, c1 = {}, c2 = {}, c3 = {};
  for (int f0 = 0; f0 < 256; f0 += 4) {
    // A[m,k] = v[b, n0+m, f0+k] = x[b, f0+k, n0+m]   (shared by all 4 tiles)
    v2f afrag;
    afrag.x = xb[(size_t)(f0 + 2 * kh)     * 256 + n0 + m];
    afrag.y = xb[(size_t)(f0 + 2 * kh + 1) * 256 + n0 + m];
    // B[k,n] = W[d0+16*t+n, wf0+f0+k] for tile t in 0..3
    const float* wp = W + (size_t)(d0 + m) * 512 + wf0 + f0 + 2 * kh;
    v2f b0 = { wp[0 * 16 * 512], wp[0 * 16 * 512 + 1] };
    v2f b1 = { wp[1 * 16 * 512], wp[1 * 16 * 512 + 1] };
    v2f b2 = { wp[2 * 16 * 512], wp[2 * 16 * 512 + 1] };
    v2f b3 = { wp[3 * 16 * 512], wp[3 * 16 * 512 + 1] };
    c0 = __builtin_amdgcn_wmma_f32_16x16x4_f32(false, afrag, false, b0, (short)0, c0, false, false);
    c1 = __builtin_amdgcn_wmma_f32_16x16x4_f32(false, afrag, false, b1, (short)0, c1, false, false);
    c2 = __builtin_amdgcn_wmma_f32_16x16x4_f32(false, afrag, false, b2, (short)0, c2, false, false);
    c3 = __builtin_amdgcn_wmma_f32_16x16x4_f32(false, afrag, false, b3, (short)0, c3, false, false);
  }

  // Branch-free epilogue: hoist b_lin (0 for R) so the store loop is pure FMA+store.
  float* dst = (isR ? Rbuf : Lbuf) + (size_t)b * 65536;
  const int ncol = lane & 15;
  const float bl0 = isR ? 0.0f : b_lin[d0 +  0 + ncol];
  const float bl1 = isR ? 0.0f : b_lin[d0 + 16 + ncol];
  const float bl2 = isR ? 0.0f : b_lin[d0 + 32 + ncol];
  const float bl3 = isR ? 0.0f : b_lin[d0 + 48 + ncol];
#pragma unroll
  for (int v = 0; v < 8; ++v) {
    const int row = v + 8 * kh;                         // C layout: lanes16-31 -> M+8
    float* r = dst + (size_t)(n0 + row) * 256 + d0 + ncol;
    r[ 0] = c0[v] + bl0;
    r[16] = c1[v] + bl1;
    r[32] = c2[v] + bl2;
    r[48] = c3[v] + bl3;
  }
}

// ---------------------------------------------------------------------------
// Kernel 2 (runtime-dominant VALU phase, ~134M LeakyReLU+FMA elements):
// e[b,i,j] = sum_d LeakyReLU(L[i,d]+R[j,d]) * a[d] + bias[i,j], adj mask,
// row softmax (all-masked row -> uniform 1/256). One block per (b,i), thread
// t owns column j=t; R staged through LDS in padded 32-d chunks; next chunk
// prefetched via global_prefetch_b8.
// ---------------------------------------------------------------------------
__global__ __launch_bounds__(256) void gat_scores(const float* __restrict__ Lbuf,
                                                  const float* __restrict__ Rbuf,
                                                  const float* __restrict__ avec,
                                                  const float* __restrict__ bias,
                                                  const int*   __restrict__ adj,
                                                  float* __restrict__ attn) {
  __shared__ float Lrow[256];
  __shared__ float as[256];
  __shared__ float Rt[256 * 33];   // 256 j-rows x 32 d, stride 33 (bank-conflict pad)
  __shared__ float red[256];

  const int tid = threadIdx.x;
  const int bi  = blockIdx.x;
  const int b   = bi >> 8;
  const int i   = bi & 255;

  Lrow[tid] = Lbuf[((size_t)b * 256 + i) * 256 + tid];   // already includes b_lin
  as[tid]   = avec[tid];
  __syncthreads();

  float acc = 0.0f;
  for (int d0 = 0; d0 < 256; d0 += 32) {
    if (d0 + 32 < 256)   // prefetch next chunk of this thread's R row into L2/L0
      __builtin_prefetch(&Rbuf[((size_t)b * 256 + tid) * 256 + d0 + 32], 0, 0);
    for (int idx = tid; idx < 256 * 32; idx += 256) {
      const int j  = idx >> 5;
      const int dd = idx & 31;
      Rt[j * 33 + dd] = Rbuf[((size_t)b * 256 + j) * 256 + d0 + dd];
    }
    __syncthreads();
#pragma unroll
    for (int dd = 0; dd < 32; ++dd) {
      const float z = Lrow[d0 + dd] + Rt[tid * 33 + dd];
      const float h = z > 0.0f ? z : GAT_ALPHA * z;
      acc += h * as[d0 + dd];
    }
    __syncthreads();
  }

  float e = acc + bias[(size_t)i * 256 + tid];
  if (adj[((size_t)b * 256 + i) * 256 + tid] == 0) e = -INFINITY;

  // block-wide max
  red[tid] = e;
  __syncthreads();
  for (int s = 128; s > 0; s >>= 1) {
    if (tid < s) red[tid] = fmaxf(red[tid], red[tid + s]);
    __syncthreads();
  }
  const float rmax = red[0];
  __syncthreads();

  float outv;
  if (rmax == -INFINITY) {          // entire row masked: reference -> e=0 -> uniform
    outv = 1.0f / 256.0f;
  } else {
    const float p = __expf(e - rmax);
    red[tid] = p;
    __syncthreads();
    for (int s = 128; s > 0; s >>= 1) {
      if (tid < s) red[tid] += red[tid + s];
      __syncthreads();
    }
    outv = p / red[0];
  }
  attn[((size_t)b * 256 + i) * 256 + tid] = outv;
}

// ---------------------------------------------------------------------------
// Kernel 3: out[b,k,d] = sigmoid(sum_j attn[b,d,j]*x[b,k,j]) + x[b,k,d]
// (attn @ v, fused sigmoid + residual + output transpose). One wave per
// 16(i) x 64(k) strip: shared A-fragment (attn rows), 4 accumulators.
// ---------------------------------------------------------------------------
__global__ __launch_bounds__(32) void gat_out(const float* __restrict__ attn,
                                              const float* __restrict__ x,
                                              float* __restrict__ out) {
  const int b      = blockIdx.z;
  const int itile  = blockIdx.y;  // 0..15 : attn-row tile (output d index)
  const int kstrip = blockIdx.x;  // 0..3  : 64 output-k columns
  const int lane   = threadIdx.x;
  const int m      = lane & 15;
  const int kh     = lane >> 4;

  const int i0 = itile * 16;
  const int k0 = kstrip * 64;
  const float* Ab = attn + (size_t)b * 65536;
  const float* xb = x    + (size_t)b * 65536;

  v8f c0 = {}, c1 = {}, c2 = {}, c3 = {};
  for (int j0 = 0; j0 < 256; j0 += 4) {
    v2f afrag;                                    // attn[i0+m, j0+k] (reused 4x)
    afrag.x = Ab[(size_t)(i0 + m) * 256 + j0 + 2 * kh];
    afrag.y = Ab[(size_t)(i0 + m) * 256 + j0 + 2 * kh + 1];
    const float* xp = xb + (size_t)(k0 + m) * 256 + j0 + 2 * kh;
    v2f b0 = { xp[0 * 16 * 256], xp[0 * 16 * 256 + 1] };
    v2f b1 = { xp[1 * 16 * 256], xp[1 * 16 * 256 + 1] };
    v2f b2 = { xp[2 * 16 * 256], xp[2 * 16 * 256 + 1] };
    v2f b3 = { xp[3 * 16 * 256], xp[3 * 16 * 256 + 1] };
    c0 = __builtin_amdgcn_wmma_f32_16x16x4_f32(false, afrag, false, b0, (short)0, c0, false, false);
    c1 = __builtin_amdgcn_wmma_f32_16x16x4_f32(false, afrag, false, b1, (short)0, c1, false, false);
    c2 = __builtin_amdgcn_wmma_f32_16x16x4_f32(false, afrag, false, b2, (short)0, c2, false, false);
    c3 = __builtin_amdgcn_wmma_f32_16x16x4_f32(false, afrag, false, b3, (short)0, c3, false, false);
  }

  const int ncol = lane & 15;
#pragma unroll
  for (int v = 0; v < 8; ++v) {
    const int row = v + 8 * kh;                    // output d index = i0+row
    v8f* dummy;
    (void)dummy;
    const float vals[4] = { c0[v], c1[v], c2[v], c3[v] };
#pragma unroll
    for (int t = 0; t < 4; ++t) {
      const int kk = k0 + t * 16 + ncol;
      const float sg = 1.0f / (1.0f + __expf(-vals[t]));
      const size_t off = (size_t)b * 65536 + (size_t)kk * 256 + (i0 + row);
      out[off] = sg + xb[(size_t)kk * 256 + (i0 + row)];
    }
  }
}

// ---------------------------------------------------------------------------
extern "C" void kernel_launch(void* const* d_in, const int* in_sizes, int n_in,
                              void* d_out, int out_size, void* d_ws, size_t ws_size,
                              hipStream_t stream) {
  const float* x     = (const float*)d_in[0];
  const int*   adj   = (const int*)  d_in[1];
  const float* W     = (const float*)d_in[2];
  const float* b_lin = (const float*)d_in[3];
  const float* avec  = (const float*)d_in[4];
  const float* bias  = (const float*)d_in[5];
  float* out = (float*)d_out;

  float* Lbuf = (float*)d_ws;                                // 8*256*256 f32 = 2MB
  float* Rbuf = Lbuf + (size_t)BATCH * 256 * 256;            // 2MB
  float* attn = Rbuf + (size_t)BATCH * 256 * 256;            // 2MB

  gat_proj  <<<dim3(8, 16, BATCH), 32,  0, stream>>>(x, W, b_lin, Lbuf, Rbuf);
  gat_scores<<<dim3(BATCH * 256),  256, 0, stream>>>(Lbuf, Rbuf, avec, bias, adj, attn);
  gat_out   <<<dim3(4, 16, BATCH), 32,  0, stream>>>(attn, x, out);
}